// GaussianScene_50654844289294
// MI455X (gfx1250) — compile-verified
//
#include <hip/hip_runtime.h>

// ---------------------------------------------------------------------------
// MI455X (gfx1250) covariance-from-quaternion kernel.  [FINAL]
//
// Roofline: 4M pts * (28B in + 36B out) = 256 MB @ 23.3 TB/s => ~11 us floor.
// ~65 FLOP/pt => ~24 TFLOP/s f32 needed, far under VALU peak. Memory bound.
// => optimize data movement:
//    - async global->LDS input staging (gfx1250 ASYNCcnt path,
//      global_load_async_to_lds_b128 + s_wait_asynccnt), NT load hint
//      (single-pass data, 256 MB total > 192 MB L2)
//    - LDS-staged, 16B-aligned, fully coalesced non-temporal b128 stores
//      (raw per-thread AoS stores would span 9 cachelines per wave instr)
// WMMA deliberately unused: per-point 3x3 products are <18% utilized in a
// 16x16x4 f32 tile, need undocumented dense-f32 B layout, and any layout-
// safe formulation costs more VALU+DS marshalling than the 12 FMAs it
// replaces. Compute is not the bottleneck; HBM is.
// ---------------------------------------------------------------------------

#define TILE 256
#define CPOL_NT 1   // gfx12+ CPol: TH[2:0]=1 (non-temporal), scope=0

typedef float v4f __attribute__((ext_vector_type(4)));

#if defined(__gfx1250__) &&                                            \
    __has_builtin(__builtin_amdgcn_global_load_async_to_lds_b128) &&   \
    __has_builtin(__builtin_amdgcn_s_wait_asynccnt)
#define USE_ASYNC 1
// Builtin signature (probe-confirmed via hipcc diagnostic): int4* parameters.
typedef int v4i __attribute__((ext_vector_type(4)));
typedef __attribute__((address_space(3))) v4i lds_v4i;
#else
#define USE_ASYNC 0
#endif

// cov = (R * diag(s)) * (R * diag(s))^T, COLMAP qvec2rotmat convention,
// quaternion NOT normalized (matches reference). o[9] row-major 3x3.
__device__ __forceinline__ void quat_scale_cov(float w, float x, float y, float z,
                                               float s0, float s1, float s2,
                                               float* __restrict__ o) {
    const float xx = x * x, yy = y * y, zz = z * z;
    const float xy = x * y, xz = x * z, yz = y * z;
    const float wx = w * x, wy = w * y, wz = w * z;

    const float r00 = 1.0f - 2.0f * (yy + zz);
    const float r01 = 2.0f * (xy - wz);
    const float r02 = 2.0f * (xz + wy);
    const float r10 = 2.0f * (xy + wz);
    const float r11 = 1.0f - 2.0f * (xx + zz);
    const float r12 = 2.0f * (yz - wx);
    const float r20 = 2.0f * (xz - wy);
    const float r21 = 2.0f * (yz + wx);
    const float r22 = 1.0f - 2.0f * (xx + yy);

    // M = R * diag(s) (same op order as reference: scale, then M M^T)
    const float m00 = r00 * s0, m01 = r01 * s1, m02 = r02 * s2;
    const float m10 = r10 * s0, m11 = r11 * s1, m12 = r12 * s2;
    const float m20 = r20 * s0, m21 = r21 * s1, m22 = r22 * s2;

    const float c00 = m00 * m00 + m01 * m01 + m02 * m02;
    const float c01 = m00 * m10 + m01 * m11 + m02 * m12;
    const float c02 = m00 * m20 + m01 * m21 + m02 * m22;
    const float c11 = m10 * m10 + m11 * m11 + m12 * m12;
    const float c12 = m10 * m20 + m11 * m21 + m12 * m22;
    const float c22 = m20 * m20 + m21 * m21 + m22 * m22;

    o[0] = c00; o[1] = c01; o[2] = c02;
    o[3] = c01; o[4] = c11; o[5] = c12;
    o[6] = c02; o[7] = c12; o[8] = c22;
}

__global__ __launch_bounds__(TILE) void GaussianCov_kernel(
    const float* __restrict__ q,   // [N,4] (w,x,y,z)
    const float* __restrict__ s,   // [N,3]
    float* __restrict__ out,       // [N,3,3]
    int N) {
    const int t        = threadIdx.x;
    const int tileBase = blockIdx.x * TILE;

    __shared__ __align__(16) float lds_q[TILE * 4];   // 4 KB
    __shared__ __align__(16) float lds_s[TILE * 3];   // 3 KB
    __shared__ __align__(16) float lds_o[TILE * 9];   // 9 KB

    const bool fullTile = (tileBase + TILE) <= N;

    if (fullTile) {   // block-uniform branch: barriers inside are legal
        float qw, qx, qy, qz, s0, s1, s2;
#if USE_ASYNC
        // --- async global -> LDS staging (tracked by ASYNCcnt), NT loads ---
        // quaternions: 256 lanes x 16B (B128), 16B-aligned
        __builtin_amdgcn_global_load_async_to_lds_b128(
            (v4i*)(q + (size_t)(tileBase + t) * 4),
            (lds_v4i*)(&lds_q[t * 4]), 0, CPOL_NT);
        // scales: 256*12B = 3072B tile = 192 lanes x 16B (B128), 16B-aligned
        if (t < (TILE * 3) / 4) {
            __builtin_amdgcn_global_load_async_to_lds_b128(
                (v4i*)(s + (size_t)tileBase * 3 + (size_t)t * 4),
                (lds_v4i*)(&lds_s[t * 4]), 0, CPOL_NT);
        }
        __builtin_amdgcn_s_wait_asynccnt(0);   // this wave's asyncs done
        __syncthreads();                       // all waves' asyncs done
        {
            const v4f qv = *(const v4f*)(&lds_q[t * 4]);
            qw = qv.x; qx = qv.y; qy = qv.z; qz = qv.w;
        }
        s0 = lds_s[t * 3 + 0];
        s1 = lds_s[t * 3 + 1];
        s2 = lds_s[t * 3 + 2];
#else
        {
            const v4f qv = *(const v4f*)(q + (size_t)(tileBase + t) * 4);
            qw = qv.x; qx = qv.y; qy = qv.z; qz = qv.w;
        }
        s0 = s[(size_t)(tileBase + t) * 3 + 0];
        s1 = s[(size_t)(tileBase + t) * 3 + 1];
        s2 = s[(size_t)(tileBase + t) * 3 + 2];
        __syncthreads();   // keep barrier count uniform with async path
#endif
        // compute into LDS staging (t*9 stride: 9 coprime with 64 banks)
        quat_scale_cov(qw, qx, qy, qz, s0, s1, s2, &lds_o[t * 9]);
        __syncthreads();

        // coalesced 16B-aligned non-temporal b128 stores:
        // tile output = 256*9*4 = 9216 B (16B-aligned), 576 v4f chunks.
        v4f* __restrict__ o4       = (v4f*)(out + (size_t)tileBase * 9);
        const v4f* __restrict__ l4 = (const v4f*)lds_o;
        #pragma unroll
        for (int i = t; i < (TILE * 9) / 4; i += TILE) {
            __builtin_nontemporal_store(l4[i], &o4[i]);
        }
    } else {
        // tail (never taken for N=4,000,000): simple scalar path, no barriers
        const int n = tileBase + t;
        if (n < N) {
            const v4f qv = *(const v4f*)(q + (size_t)n * 4);
            const float s0 = s[(size_t)n * 3 + 0];
            const float s1 = s[(size_t)n * 3 + 1];
            const float s2 = s[(size_t)n * 3 + 2];
            float o[9];
            quat_scale_cov(qv.x, qv.y, qv.z, qv.w, s0, s1, s2, o);
            #pragma unroll
            for (int i = 0; i < 9; ++i) out[(size_t)n * 9 + i] = o[i];
        }
    }
}

extern "C" void kernel_launch(void* const* d_in, const int* in_sizes, int n_in,
                              void* d_out, int out_size, void* d_ws, size_t ws_size,
                              hipStream_t stream) {
    (void)n_in; (void)out_size; (void)d_ws; (void)ws_size;
    const float* q = (const float*)d_in[0];   // [N,4] f32
    const float* s = (const float*)d_in[1];   // [N,3] f32
    float* out     = (float*)d_out;           // [N,9] f32
    const int N      = in_sizes[0] / 4;
    const int blocks = (N + TILE - 1) / TILE;
    GaussianCov_kernel<<<dim3(blocks), dim3(TILE), 0, stream>>>(q, s, out, N);
}